// BahdanauAttention_2456721293846
// MI455X (gfx1250) — compile-verified
//
#include <hip/hip_runtime.h>
#include <math.h>

// Problem constants (from reference)
#define TQ_ 256
#define TV_ 1024
#define NB_ 4
#define D_  128
#define U_  128
#define NEG_INF_ (-1e9f)

typedef __attribute__((ext_vector_type(2))) float v2f;
typedef __attribute__((ext_vector_type(8))) float v8f;

#if __has_builtin(__builtin_amdgcn_tanhf)
#define FAST_TANH(x) __builtin_amdgcn_tanhf(x)
#else
#define FAST_TANH(x) tanhf(x)
#endif

// ---------------------------------------------------------------------------
// Kernel 1: projection GEMM.  OUT[b][t][u] = sum_d X[t][b][d] * W[d][u]
//   X: time-major [T, NB_, D_], W: [D_, U_], OUT: [NB_*T, U_] (b-major)
// One wave computes a 16x16 output tile via V_WMMA_F32_16X16X4_F32,
// looping K over D_ in steps of 4.  Block = 256 threads = 8 waves = 8 U-tiles.
// ---------------------------------------------------------------------------
__global__ __launch_bounds__(256)
void proj_gemm_wmma(const float* __restrict__ X, const float* __restrict__ W,
                    float* __restrict__ OUT, int T) {
    const int tid   = threadIdx.x;
    const int lane  = tid & 31;
    const int wave  = tid >> 5;
    const int tilesPerB = T >> 4;
    const int bb    = blockIdx.x / tilesPerB;
    const int tbase = (blockIdx.x % tilesPerB) << 4;

    // WMMA f32 16x16x4 lane mapping (wave32):
    //   A: lane holds row m=lane&15, cols {koff, koff+1}, koff = lane<16 ? 0 : 2
    //   B: lane holds col n=lane&15, rows {koff, koff+1}
    //   C/D: c[r] is row (r + (lane<16?0:8)), col n
    const int n    = lane & 15;
    const int m    = lane & 15;
    const int koff = (lane < 16) ? 0 : 2;
    const int ubase = wave << 4;

    const float* arow = X + ((size_t)(tbase + m) * NB_ + bb) * D_;

    v8f c = {};
    for (int k = 0; k < D_; k += 4) {
        v2f a = *(const v2f*)(arow + k + koff);            // contiguous pair
        v2f bm;
        bm.x = W[(size_t)(k + koff)     * U_ + ubase + n];
        bm.y = W[(size_t)(k + koff + 1) * U_ + ubase + n];
        c = __builtin_amdgcn_wmma_f32_16x16x4_f32(false, a, false, bm,
                                                  (short)0, c, false, false);
    }

    float* orow = OUT + ((size_t)(bb * T + tbase)) * U_ + ubase + n;
    const int mbase = (lane < 16) ? 0 : 8;
#pragma unroll
    for (int r = 0; r < 8; ++r)
        orow[(size_t)(r + mbase) * U_] = c[r];
}

// ---------------------------------------------------------------------------
// Kernel 2: fused additive-score + online softmax + ctx (flash style).
// One workgroup (256 thr = 8 waves) owns (batch bb, 16 query rows).
// Loop over Tv in chunks of 128:
//   phase 1: scores chunk [16][128] -> LDS   (tanh-heavy VALU work)
//   phase 2: per-row online softmax stats, probs written back to LDS
//   phase 3: rescale WMMA accumulators, ctx += P(16x128) * V(128x128)
//            via 32x V_WMMA_F32_16X16X4_F32 per wave (wave = 16-wide D tile)
// ---------------------------------------------------------------------------
__global__ __launch_bounds__(256)
void bahdanau_attn_wmma(const float* __restrict__ value,   // [TV_, NB_, D_]
                        const unsigned char* __restrict__ mask, // [TV_, NB_]
                        const float* __restrict__ scale,   // [U_]
                        const float* __restrict__ wq,      // [NB_, TQ_, U_]
                        const float* __restrict__ wk,      // [NB_, TV_, U_]
                        float* __restrict__ out) {         // [TQ_, NB_, D_]
    __shared__ float wq_s[16][U_];     // 8 KB
    __shared__ float sc[16][128];      // 8 KB  (scores -> probs per chunk)
    __shared__ float scale_s[U_];      // 0.5 KB
    __shared__ float m_row[16], l_row[16], f_row[16];

    const int tid   = threadIdx.x;
    const int lane  = tid & 31;
    const int wave  = tid >> 5;
    const int bb    = blockIdx.x >> 4;          // TQ_/16 == 16 tiles per batch
    const int qbase = (blockIdx.x & 15) << 4;

    for (int i = tid; i < 16 * U_; i += 256) {
        const int q = i >> 7, u = i & (U_ - 1);
        wq_s[q][u] = wq[((size_t)(bb * TQ_ + qbase + q)) * U_ + u];
    }
    if (tid < U_) scale_s[tid] = scale[tid];
    if (tid < 16) { m_row[tid] = -3.0e38f; l_row[tid] = 0.f; f_row[tid] = 1.f; }
    __syncthreads();

    // WMMA lane constants (see kernel 1)
    const int n     = lane & 15;
    const int mA    = lane & 15;
    const int koff  = (lane < 16) ? 0 : 2;
    const int mbase = (lane < 16) ? 0 : 8;
    const int dbase = wave << 4;

    v8f cacc = {};

    // phase-1 work split: 2048 (q,v) pairs -> 128 v-threads x 2 q-groups of 8
    const int vloc = tid & 127;
    const int qoff = (tid >> 7) << 3;           // 0 or 8

    for (int chunk = 0; chunk < TV_ / 128; ++chunk) {
        const int vbase = chunk * 128;

        // ---- phase 1: raw additive scores -> LDS ----
        {
            float acc[8];
#pragma unroll
            for (int j = 0; j < 8; ++j) acc[j] = 0.f;
            const float* wkrow = wk + ((size_t)(bb * TV_ + vbase + vloc)) * U_;
            for (int u = 0; u < U_; u += 4) {
                const float4 k4 = *(const float4*)(wkrow + u);
                const float4 s4 = *(const float4*)(&scale_s[u]);
#pragma unroll
                for (int j = 0; j < 8; ++j) {
                    const float4 q4 = *(const float4*)(&wq_s[qoff + j][u]);
                    acc[j] += s4.x * FAST_TANH(q4.x + k4.x)
                            + s4.y * FAST_TANH(q4.y + k4.y)
                            + s4.z * FAST_TANH(q4.z + k4.z)
                            + s4.w * FAST_TANH(q4.w + k4.w);
                }
            }
            const bool valid = mask[(size_t)(vbase + vloc) * NB_ + bb] != 0;
#pragma unroll
            for (int j = 0; j < 8; ++j)
                sc[qoff + j][vloc] = valid ? acc[j] : NEG_INF_;
        }
        __syncthreads();

        // ---- phase 2: online softmax; wave w handles rows w and w+8 ----
#pragma unroll
        for (int rr = 0; rr < 2; ++rr) {
            const int q = wave + rr * 8;
            float mx = -3.0e38f;
#pragma unroll
            for (int j = 0; j < 4; ++j) mx = fmaxf(mx, sc[q][lane + 32 * j]);
#pragma unroll
            for (int off = 16; off > 0; off >>= 1)
                mx = fmaxf(mx, __shfl_xor(mx, off, 32));
            const float mprev = m_row[q];
            const float mnew  = fmaxf(mprev, mx);
            const float f     = __expf(mprev - mnew);
            float s = 0.f;
#pragma unroll
            for (int j = 0; j < 4; ++j) {
                const float e = __expf(sc[q][lane + 32 * j] - mnew);
                sc[q][lane + 32 * j] = e;
                s += e;
            }
#pragma unroll
            for (int off = 16; off > 0; off >>= 1)
                s += __shfl_xor(s, off, 32);
            if (lane == 0) {
                m_row[q] = mnew;
                l_row[q] = l_row[q] * f + s;
                f_row[q] = f;
            }
        }
        __syncthreads();

        // ---- phase 3: rescale accumulators, ctx += P * V via WMMA ----
#pragma unroll
        for (int r = 0; r < 8; ++r) cacc[r] *= f_row[r + mbase];

        const float* vb = value + ((size_t)vbase * NB_ + bb) * D_ + dbase + n;
        for (int k = 0; k < 128; k += 4) {
            v2f a = *(const v2f*)(&sc[mA][k + koff]);       // probs from LDS
            v2f bm;
            bm.x = vb[(size_t)(k + koff)     * (NB_ * D_)];
            bm.y = vb[(size_t)(k + koff + 1) * (NB_ * D_)];
            cacc = __builtin_amdgcn_wmma_f32_16x16x4_f32(false, a, false, bm,
                                                         (short)0, cacc, false, false);
        }
        __syncthreads();   // protect sc before next chunk's phase 1
    }

    // ---- epilogue: normalize by softmax denominator, store time-major ----
#pragma unroll
    for (int r = 0; r < 8; ++r) {
        const int mo = r + mbase;
        const float invl = 1.f / l_row[mo];
        out[((size_t)(qbase + mo) * NB_ + bb) * D_ + dbase + n] = cacc[r] * invl;
    }
}

// ---------------------------------------------------------------------------
extern "C" void kernel_launch(void* const* d_in, const int* in_sizes, int n_in,
                              void* d_out, int out_size, void* d_ws, size_t ws_size,
                              hipStream_t stream) {
    (void)in_sizes; (void)n_in; (void)out_size; (void)ws_size;

    const float*         query = (const float*)d_in[0];        // [TQ_, NB_, D_]
    const float*         value = (const float*)d_in[1];        // [TV_, NB_, D_]
    const unsigned char* mask  = (const unsigned char*)d_in[2];// [TV_, NB_] bool
    const float*         W1    = (const float*)d_in[3];        // [D_, U_]
    const float*         W2    = (const float*)d_in[4];        // [D_, U_]
    const float*         scale = (const float*)d_in[5];        // [U_]
    float*               out   = (float*)d_out;                // [TQ_, NB_, D_]

    float* wq_ws = (float*)d_ws;                               // NB_*TQ_*U_ f32
    float* wk_ws = wq_ws + (size_t)NB_ * TQ_ * U_;             // NB_*TV_*U_ f32

    // wq = q @ W1, wk = v @ W2  (b-major outputs for the attention kernel)
    proj_gemm_wmma<<<(NB_ * TQ_) / 16, 256, 0, stream>>>(query, W1, wq_ws, TQ_);
    proj_gemm_wmma<<<(NB_ * TV_) / 16, 256, 0, stream>>>(value, W2, wk_ws, TV_);

    // fused scores + masked online softmax + ctx
    bahdanau_attn_wmma<<<NB_ * (TQ_ / 16), 256, 0, stream>>>(
        value, mask, scale, wq_ws, wk_ws, out);
}